// DModel_46557445488745
// MI455X (gfx1250) — compile-verified
//
#include <hip/hip_runtime.h>

typedef float v2f __attribute__((ext_vector_type(2)));
typedef float v8f __attribute__((ext_vector_type(8)));

#define TT    500
#define HH    40
#define PITCH 50      // even (b64-aligned A loads), 50 mod 64 banking is conflict-friendly
#define WAVES 8
#define ROWS_PER_BLOCK (WAVES * 16)

static __device__ __forceinline__ float fast_tanh(float x) {
    float y;
    asm("v_tanh_f32 %0, %1" : "=v"(y) : "v"(x));
    return y;
}

// Compiler-only fence: keeps DS program order across the read/write phases.
// Hardware executes same-wave DS ops in order, so no s_barrier is needed for
// the wave-private state tile.
static __device__ __forceinline__ void lds_order_fence() {
    asm volatile("" ::: "memory");
}

// Fused weight matrix: rows 0..39 = kernel_hidden, 40..41 = kernel_state,
// 42 = kernel_action, 43 = bias. Columns >= 40 are zero padding.
static __device__ __forceinline__ float wfull(int r, int c,
                                              const float* __restrict__ Wh,
                                              const float* __restrict__ Ks,
                                              const float* __restrict__ Ka,
                                              const float* __restrict__ bias) {
    if (c >= HH) return 0.0f;
    if (r < HH)  return Wh[r * HH + c];
    if (r == HH)     return Ks[c];
    if (r == HH + 1) return Ks[HH + c];
    if (r == HH + 2) return Ka[c];
    return bias[c];
}

__global__ __launch_bounds__(256, 1)
void rnn_scan_wmma(const float* __restrict__ x,
                   const float* __restrict__ Ks,
                   const float* __restrict__ Wh,
                   const float* __restrict__ Ka,
                   const float* __restrict__ bias,
                   const float* __restrict__ Dw,
                   const float* __restrict__ Db,
                   float* __restrict__ out)
{
    __shared__ float s2s[WAVES][16][PITCH];   // wave-private 16x48 state tiles

    const int tid     = threadIdx.x;
    const int w       = tid >> 5;
    const int lane    = tid & 31;
    const int m       = lane & 15;        // batch row within wave tile
    const int half    = lane >> 4;        // lane half selects K-offset (+2)
    const int halfoff = half * 2;

    // ---- zero-init own wave's state tile (no cross-wave traffic) ----
    {
        float* myz = (float*)&s2s[w][0][0];
        for (int i = lane; i < 16 * PITCH; i += 32) myz[i] = 0.0f;
        lds_order_fence();
    }

    // ---- load fused weights into registers in WMMA B-layout ----
    // B chunk (4K x 16N): VGPR0 = row K0+2*half, VGPR1 = row K0+1+2*half, col = n*16+m
    v2f wv[11][3];
    #pragma unroll
    for (int ck = 0; ck < 11; ++ck) {
        #pragma unroll
        for (int n = 0; n < 3; ++n) {
            const int col = n * 16 + m;
            const int r0  = ck * 4 + halfoff;
            wv[ck][n].x = wfull(r0,     col, Wh, Ks, Ka, bias);
            wv[ck][n].y = wfull(r0 + 1, col, Wh, Ks, Ka, bias);
        }
    }

    // ---- per-lane x streaming (pointer-increment, 1-step prefetch) ----
    const int    brow = blockIdx.x * ROWS_PER_BLOCK + w * 16 + m;
    const float* pxt  = x + (size_t)brow * TT * 3;
    const int    f0   = 2 * half;   // lane half 0 -> feature 0/1, half 1 -> feature 2/const
    const float  mean0   = half ? 0.0f : -0.3f;
    const float  iscale0 = half ? 1.0f : (1.0f / 0.9f);
    float p0 = pxt[f0];
    float p1 = pxt[1];

    const float* sbase = &s2s[w][m][0];
    float*       dstb  = &s2s[w][8 * half][m];

    for (int t = 0; t < TT; ++t) {
        // ---- gather ALL A-chunks first: 5 fused ds_load_2addr_b64, one wait ----
        v2f a[11];
        #pragma unroll
        for (int ck = 0; ck < 10; ++ck)
            a[ck] = *(const v2f*)(sbase + ck * 4 + halfoff);

        // A chunk 10 from registers: [xn0 xn1 | xn2 1]
        a[10].x = (p0 - mean0) * iscale0;
        a[10].y = half ? 1.0f : p1 * (1.0f / 0.07f);

        // prefetch next step's x (clamped pointer, no OOB)
        const float* pn = (t < TT - 1) ? (pxt + 3) : pxt;
        p0  = pn[f0];
        p1  = pn[1];
        pxt = pn;

        lds_order_fence();   // stores below must not move above the loads

        // ---- 33 back-to-back WMMAs: [s2 | xn | 1](16x44) @ Wfull(44x48) ----
        v8f acc0 = {}, acc1 = {}, acc2 = {};
        #pragma unroll
        for (int ck = 0; ck < 11; ++ck) {
            acc0 = __builtin_amdgcn_wmma_f32_16x16x4_f32(false, a[ck], false, wv[ck][0],
                                                         (short)0, acc0, false, false);
            acc1 = __builtin_amdgcn_wmma_f32_16x16x4_f32(false, a[ck], false, wv[ck][1],
                                                         (short)0, acc1, false, false);
            acc2 = __builtin_amdgcn_wmma_f32_16x16x4_f32(false, a[ck], false, wv[ck][2],
                                                         (short)0, acc2, false, false);
        }

        // ---- tanh (batched: consecutive independent TRANS ops) ----
        float tv[24];
        #pragma unroll
        for (int r = 0; r < 8; ++r) {
            tv[r]      = fast_tanh(acc0[r]);
            tv[8 + r]  = fast_tanh(acc1[r]);
            tv[16 + r] = fast_tanh(acc2[r]);
        }

        // ---- store new state: D row M = r + 8*half, col N = n*16 + m ----
        // constant offsets from one base -> compiler can fuse ds_store_2addr_b32
        #pragma unroll
        for (int r = 0; r < 8; ++r) {
            dstb[r * PITCH +  0] = tv[r];
            dstb[r * PITCH + 16] = tv[8 + r];
            dstb[r * PITCH + 32] = tv[16 + r];
        }

        lds_order_fence();   // next iteration's loads must not move above stores
    }

    // ---- dense head (wave-private, no barrier): 16 rows x 2 cols = 32 lanes ----
    {
        lds_order_fence();
        const int orow = lane >> 1;     // 0..15
        const int oc   = lane & 1;
        const float* srow = &s2s[w][orow][0];
        float sum = Db[oc];
        #pragma unroll
        for (int h = 0; h < HH; ++h)
            sum += srow[h] * Dw[h * 2 + oc];
        const float so = oc ? 0.07f : 0.9f;
        const float mo = oc ? 0.0f  : -0.3f;
        float v = sum * so + mo;
        const float lo = oc ? -0.07f : -1.2f;
        const float hi = oc ?  0.07f :  0.6f;
        v = fminf(fmaxf(v, lo), hi);
        out[((size_t)(blockIdx.x * ROWS_PER_BLOCK + w * 16 + orow)) * 2 + oc] = v;
    }
}

extern "C" void kernel_launch(void* const* d_in, const int* in_sizes, int n_in,
                              void* d_out, int out_size, void* d_ws, size_t ws_size,
                              hipStream_t stream) {
    const float* x    = (const float*)d_in[0];
    const float* Ks   = (const float*)d_in[1];
    const float* Wh   = (const float*)d_in[2];
    const float* Ka   = (const float*)d_in[3];
    const float* bias = (const float*)d_in[4];
    const float* Dw   = (const float*)d_in[5];
    const float* Db   = (const float*)d_in[6];
    float* out = (float*)d_out;

    const int B = in_sizes[0] / (TT * 3);           // 32768
    dim3 grid(B / ROWS_PER_BLOCK), block(256);
    rnn_scan_wmma<<<grid, block, 0, stream>>>(x, Ks, Wh, Ka, bias, Dw, Db, out);
}